// LinearAttention_17016660427057
// MI455X (gfx1250) — compile-verified
//
#include <hip/hip_runtime.h>
#include <math.h>

typedef __attribute__((ext_vector_type(2))) float v2f;
typedef __attribute__((ext_vector_type(8))) float v8f;

#define B_   32
#define C_   128
#define L_   8192
#define H_   4
#define D_   32
#define SCALE_ 0.17677669529663687f   // 32^-0.5
#define LT   32                        // l-tile per block for GEMM kernels
#define XS   40                        // padded LDS row stride (2-row offset = 80 ≡ 16 mod 64 banks)

__device__ __forceinline__ v8f wmma_f32(v2f a, v2f b, v8f c) {
    // D = A(16x4 f32) * B(4x16 f32) + C(16x16 f32)
    return __builtin_amdgcn_wmma_f32_16x16x4_f32(
        /*neg_a=*/false, a, /*neg_b=*/false, b,
        /*c_mod=*/(short)0, c, /*reuse_a=*/false, /*reuse_b=*/false);
}

// ---------------------------------------------------------------------------
// K1: qkv = w_qkv[384x128] * x[b,128,Ltile]; q gets softmax over d (32 rows/head)
// grid (L/LT, B), 256 threads (8 waves)
// ---------------------------------------------------------------------------
__global__ __launch_bounds__(256) void k_qkv(
    const float* __restrict__ x, const float* __restrict__ wqkv,
    float* __restrict__ qbuf, float* __restrict__ kbuf, float* __restrict__ vbuf)
{
    __shared__ float xt[128 * XS];
    __shared__ float qt[128 * XS];
    const int tid  = threadIdx.x;
    const int lane = tid & 31;
    const int wv   = tid >> 5;
    const int bb   = blockIdx.y;
    const size_t l0 = (size_t)blockIdx.x * LT;

    // stage x tile [128][32] (float4, coalesced)
    {
        const float* xb = x + ((size_t)bb * C_) * L_ + l0;
        #pragma unroll
        for (int i = 0; i < 4; ++i) {
            int idx = tid + i * 256;          // 0..1023
            int row = idx >> 3;               // 8 float4 per row
            int c4  = (idx & 7) << 2;
            float4 v4 = *(const float4*)(xb + (size_t)row * L_ + c4);
            float* dst = &xt[row * XS + c4];
            dst[0] = v4.x; dst[1] = v4.y; dst[2] = v4.z; dst[3] = v4.w;
        }
    }
    __syncthreads();

    const int r15   = lane & 15;
    const int hi    = lane >> 4;      // 0/1: lane half
    const int kb    = hi * 2;         // K sub-offset for A/B fragments
    const int mrow0 = hi << 3;        // +8 rows for upper lane half in C/D

    for (int t = wv; t < 48; t += 8) {          // 24 M-tiles x 2 N-tiles
        const int mt = t >> 1, nt = t & 1;
        v8f acc = {0.f,0.f,0.f,0.f,0.f,0.f,0.f,0.f};
        const float* wr = wqkv + (size_t)(mt * 16 + r15) * C_;
        for (int k = 0; k < C_; k += 4) {
            v2f a, b;
            a[0] = wr[k + kb];
            a[1] = wr[k + kb + 1];
            b[0] = xt[(k + kb)     * XS + nt * 16 + r15];
            b[1] = xt[(k + kb + 1) * XS + nt * 16 + r15];
            acc = wmma_f32(a, b, acc);
        }
        if (mt < 8) {                                   // q rows 0..127 -> LDS
            #pragma unroll
            for (int v = 0; v < 8; ++v) {
                int row = mt * 16 + mrow0 + v;
                qt[row * XS + nt * 16 + r15] = acc[v];
            }
        } else {                                        // k/v rows -> global
            float* dst  = (mt < 16) ? kbuf : vbuf;
            int obase   = (mt < 16) ? (mt - 8) * 16 : (mt - 16) * 16;
            #pragma unroll
            for (int v = 0; v < 8; ++v) {
                int row = obase + mrow0 + v;
                dst[((size_t)bb * C_ + row) * L_ + l0 + nt * 16 + r15] = acc[v];
            }
        }
    }
    __syncthreads();

    // q softmax over d=32 per (head, column), then *SCALE
    if (tid < H_ * LT) {
        int h = tid >> 5, n = tid & 31;
        float m = -3.4e38f;
        #pragma unroll
        for (int d = 0; d < 32; ++d) m = fmaxf(m, qt[(h * 32 + d) * XS + n]);
        float e[32], s = 0.f;
        #pragma unroll
        for (int d = 0; d < 32; ++d) { e[d] = __expf(qt[(h * 32 + d) * XS + n] - m); s += e[d]; }
        float inv = SCALE_ / s;
        #pragma unroll
        for (int d = 0; d < 32; ++d)
            qbuf[((size_t)bb * C_ + h * 32 + d) * L_ + l0 + n] = e[d] * inv;
    }
}

// ---------------------------------------------------------------------------
// K2: per k-row (b,h,d) max and 1/sum(exp) over l. grid = B*C rows.
// ---------------------------------------------------------------------------
__global__ __launch_bounds__(256) void k_kstats(
    const float* __restrict__ kbuf, float* __restrict__ kmax, float* __restrict__ kinv)
{
    __shared__ float red[256];
    const int row = blockIdx.x;
    const int tid = threadIdx.x;
    const float* kr = kbuf + (size_t)row * L_;
    float m = -3.4e38f;
    for (int i = tid; i < L_; i += 256) m = fmaxf(m, kr[i]);
    red[tid] = m; __syncthreads();
    for (int s = 128; s > 0; s >>= 1) { if (tid < s) red[tid] = fmaxf(red[tid], red[tid + s]); __syncthreads(); }
    m = red[0]; __syncthreads();
    float sum = 0.f;
    for (int i = tid; i < L_; i += 256) sum += __expf(kr[i] - m);
    red[tid] = sum; __syncthreads();
    for (int s = 128; s > 0; s >>= 1) { if (tid < s) red[tid] += red[tid + s]; __syncthreads(); }
    if (tid == 0) { kmax[row] = m; kinv[row] = 1.f / red[0]; }
}

// ---------------------------------------------------------------------------
// K3: ctx[b,h,d,e] = sum_n softmax(k)[d,n] * v[e,n].  grid = B*H, 8 waves.
// ---------------------------------------------------------------------------
#define CH 128
#define CS 133   // padded: lane-row stride 133 -> conflict-free fragment gathers
__global__ __launch_bounds__(256) void k_ctx(
    const float* __restrict__ kbuf, const float* __restrict__ vbuf,
    const float* __restrict__ kmax, const float* __restrict__ kinv,
    float* __restrict__ ctx)
{
    __shared__ float lds[2 * 32 * CS];   // kc | vc, later reused as red[8*1024]
    __shared__ float sm[32], si[32];
    float* kc = lds;
    float* vc = lds + 32 * CS;

    const int tid  = threadIdx.x;
    const int lane = tid & 31;
    const int wv   = tid >> 5;
    const int bh   = blockIdx.x;         // b*4 + h
    const int r15   = lane & 15;
    const int hi    = lane >> 4;
    const int kb    = hi * 2;
    const int mrow0 = hi << 3;

    if (tid < 32) { sm[tid] = kmax[bh * 32 + tid]; si[tid] = kinv[bh * 32 + tid]; }

    const float* kg = kbuf + (size_t)bh * 32 * L_;
    const float* vg = vbuf + (size_t)bh * 32 * L_;

    v8f z = {0.f,0.f,0.f,0.f,0.f,0.f,0.f,0.f};
    v8f acc[4] = {z, z, z, z};           // (dt,et) in {0,1}^2

    for (int ch = 0; ch < L_; ch += CH) {
        __syncthreads();                 // protects prior-iter reads + sm/si init
        #pragma unroll
        for (int i = 0; i < 4; ++i) {
            int idx = tid + i * 256;     // 0..1023 (32 rows x 32 float4)
            int row = idx >> 5;
            int c4  = (idx & 31) << 2;
            float4 kv = *(const float4*)(kg + (size_t)row * L_ + ch + c4);
            float m = sm[row], inv = si[row];
            kc[row * CS + c4 + 0] = __expf(kv.x - m) * inv;
            kc[row * CS + c4 + 1] = __expf(kv.y - m) * inv;
            kc[row * CS + c4 + 2] = __expf(kv.z - m) * inv;
            kc[row * CS + c4 + 3] = __expf(kv.w - m) * inv;
            float4 vv = *(const float4*)(vg + (size_t)row * L_ + ch + c4);
            vc[row * CS + c4 + 0] = vv.x;
            vc[row * CS + c4 + 1] = vv.y;
            vc[row * CS + c4 + 2] = vv.z;
            vc[row * CS + c4 + 3] = vv.w;
        }
        __syncthreads();
        const int cb = wv * 16;          // this wave's 16 columns of the chunk
        for (int ks = 0; ks < 16; ks += 4) {
            const int kk = cb + ks + kb;
            #pragma unroll
            for (int tI = 0; tI < 4; ++tI) {
                int dt = tI >> 1, et = tI & 1;
                v2f a, b;
                a[0] = kc[(dt * 16 + r15) * CS + kk];
                a[1] = kc[(dt * 16 + r15) * CS + kk + 1];
                b[0] = vc[(et * 16 + r15) * CS + kk];
                b[1] = vc[(et * 16 + r15) * CS + kk + 1];
                acc[tI] = wmma_f32(a, b, acc[tI]);
            }
        }
    }

    __syncthreads();
    float* red = lds;                    // 8 waves x 1024 = 32KB (fits in lds)
    #pragma unroll
    for (int tI = 0; tI < 4; ++tI) {
        int dt = tI >> 1, et = tI & 1;
        #pragma unroll
        for (int v = 0; v < 8; ++v) {
            int d = dt * 16 + mrow0 + v;
            int e = et * 16 + r15;
            red[wv * 1024 + d * 32 + e] = acc[tI][v];
        }
    }
    __syncthreads();
    for (int j = tid; j < 1024; j += 256) {
        float s = 0.f;
        #pragma unroll
        for (int w = 0; w < 8; ++w) s += red[w * 1024 + j];
        ctx[(size_t)bh * 1024 + j] = s;
    }
}

// ---------------------------------------------------------------------------
// K4a: W2[b][o][h*32+d] = sum_e w_out[o][h*32+e] * ctx[b,h,d,e]. grid = B*H.
// ---------------------------------------------------------------------------
__global__ __launch_bounds__(256) void k_w2(
    const float* __restrict__ wout, const float* __restrict__ ctx, float* __restrict__ W2)
{
    __shared__ float cl[1024];
    const int bh = blockIdx.x;
    const int b = bh >> 2, h = bh & 3;
    const int tid = threadIdx.x;
    for (int i = tid; i < 1024; i += 256) cl[i] = ctx[(size_t)bh * 1024 + i];
    __syncthreads();
    for (int j = tid; j < 4096; j += 256) {
        int o = j >> 5, d = j & 31;
        const float* wr = wout + o * C_ + h * 32;
        float s = 0.f;
        #pragma unroll 8
        for (int e = 0; e < 32; ++e) s += wr[e] * cl[d * 32 + e];
        W2[((size_t)b * C_ + o) * C_ + h * 32 + d] = s;
    }
}

// ---------------------------------------------------------------------------
// K5: out = W2[b](128x128) * q[b](128xL) + b_out -> d_out; per-block sum/sumsq
// grid (L/LT, B), 8 waves.
// ---------------------------------------------------------------------------
__global__ __launch_bounds__(256) void k_out(
    const float* __restrict__ W2, const float* __restrict__ qbuf,
    const float* __restrict__ bout, float* __restrict__ out,
    float* __restrict__ partials)
{
    __shared__ float xt[128 * XS];
    __shared__ float red[512];
    const int tid  = threadIdx.x;
    const int lane = tid & 31;
    const int wv   = tid >> 5;
    const int bb   = blockIdx.y;
    const size_t l0 = (size_t)blockIdx.x * LT;

    {
        const float* qb = qbuf + ((size_t)bb * C_) * L_ + l0;
        #pragma unroll
        for (int i = 0; i < 4; ++i) {
            int idx = tid + i * 256;
            int row = idx >> 3;
            int c4  = (idx & 7) << 2;
            float4 v4 = *(const float4*)(qb + (size_t)row * L_ + c4);
            float* dst = &xt[row * XS + c4];
            dst[0] = v4.x; dst[1] = v4.y; dst[2] = v4.z; dst[3] = v4.w;
        }
    }
    __syncthreads();

    const int r15   = lane & 15;
    const int hi    = lane >> 4;
    const int kb    = hi * 2;
    const int mrow0 = hi << 3;

    float lsum = 0.f, lsq = 0.f;
    for (int t = wv; t < 16; t += 8) {          // 8 M-tiles x 2 N-tiles
        const int mt = t >> 1, nt = t & 1;
        v8f acc = {0.f,0.f,0.f,0.f,0.f,0.f,0.f,0.f};
        const float* wr = W2 + ((size_t)bb * C_ + mt * 16 + r15) * C_;
        for (int k = 0; k < C_; k += 4) {
            v2f a, b;
            a[0] = wr[k + kb];
            a[1] = wr[k + kb + 1];
            b[0] = xt[(k + kb)     * XS + nt * 16 + r15];
            b[1] = xt[(k + kb + 1) * XS + nt * 16 + r15];
            acc = wmma_f32(a, b, acc);
        }
        #pragma unroll
        for (int v = 0; v < 8; ++v) {
            int row = mt * 16 + mrow0 + v;
            float val = acc[v] + bout[row];
            out[((size_t)bb * C_ + row) * L_ + l0 + nt * 16 + r15] = val;
            lsum += val; lsq += val * val;
        }
    }

    red[tid] = lsum; red[256 + tid] = lsq;
    __syncthreads();
    for (int s = 128; s > 0; s >>= 1) {
        if (tid < s) { red[tid] += red[tid + s]; red[256 + tid] += red[256 + tid + s]; }
        __syncthreads();
    }
    if (tid == 0) {
        partials[((size_t)bb * 256 + blockIdx.x) * 2 + 0] = red[0];
        partials[((size_t)bb * 256 + blockIdx.x) * 2 + 1] = red[256];
    }
}

// ---------------------------------------------------------------------------
// K6: per-batch mean + rsqrt(var+eps) from 256 partial pairs. grid = B.
// ---------------------------------------------------------------------------
__global__ __launch_bounds__(256) void k_gnstat(
    const float* __restrict__ partials, float* __restrict__ gns)
{
    __shared__ float red[512];
    const int b = blockIdx.x, tid = threadIdx.x;
    red[tid]       = partials[((size_t)b * 256 + tid) * 2 + 0];
    red[256 + tid] = partials[((size_t)b * 256 + tid) * 2 + 1];
    __syncthreads();
    for (int s = 128; s > 0; s >>= 1) {
        if (tid < s) { red[tid] += red[tid + s]; red[256 + tid] += red[256 + tid + s]; }
        __syncthreads();
    }
    if (tid == 0) {
        const float N = (float)C_ * (float)L_;
        float mean = red[0] / N;
        float var  = red[256] / N - mean * mean;
        gns[b * 2 + 0] = mean;
        gns[b * 2 + 1] = rsqrtf(var + 1e-5f);
    }
}

// ---------------------------------------------------------------------------
// K7: in-place normalize d_out: (v-mean)*inv*gn_w[c] + gn_b[c]. float4 grid.
// ---------------------------------------------------------------------------
__global__ __launch_bounds__(256) void k_gnorm(
    float* __restrict__ out, const float* __restrict__ gns,
    const float* __restrict__ gnw, const float* __restrict__ gnb)
{
    size_t i = ((size_t)blockIdx.x * 256 + threadIdx.x) * 4;
    int b = (int)(i >> 20);              // 128*8192 = 2^20 per batch
    int c = (int)((i >> 13) & 127);      // 8192 = 2^13 per channel row
    float mean = gns[b * 2], inv = gns[b * 2 + 1];
    float w = gnw[c] * inv, bias = gnb[c];
    float4 v = *(float4*)(out + i);
    v.x = (v.x - mean) * w + bias;
    v.y = (v.y - mean) * w + bias;
    v.z = (v.z - mean) * w + bias;
    v.w = (v.w - mean) * w + bias;
    *(float4*)(out + i) = v;
}

// ---------------------------------------------------------------------------
extern "C" void kernel_launch(void* const* d_in, const int* in_sizes, int n_in,
                              void* d_out, int out_size, void* d_ws, size_t ws_size,
                              hipStream_t stream) {
    const float* x    = (const float*)d_in[0];
    const float* wqkv = (const float*)d_in[1];
    const float* wout = (const float*)d_in[2];
    const float* bout = (const float*)d_in[3];
    const float* gnw  = (const float*)d_in[4];
    const float* gnb  = (const float*)d_in[5];
    float* out = (float*)d_out;

    const size_t PLANE = (size_t)B_ * C_ * L_;   // 33,554,432 floats
    float* ws    = (float*)d_ws;
    float* qbuf  = ws;
    float* kbuf  = qbuf + PLANE;
    float* vbuf  = kbuf + PLANE;
    float* kmax  = vbuf + PLANE;                 // B*C = 4096
    float* kinv  = kmax + (size_t)B_ * C_;       // 4096
    float* ctx   = kinv + (size_t)B_ * C_;       // B*H*32*32 = 131072
    float* W2    = ctx + (size_t)B_ * H_ * 1024; // B*128*128 = 524288
    float* parts = W2 + (size_t)B_ * C_ * C_;    // B*256*2 = 16384
    float* gns   = parts + (size_t)B_ * 256 * 2; // 64

    k_qkv   <<<dim3(L_ / LT, B_), 256, 0, stream>>>(x, wqkv, qbuf, kbuf, vbuf);
    k_kstats<<<B_ * C_,           256, 0, stream>>>(kbuf, kmax, kinv);
    k_ctx   <<<B_ * H_,           256, 0, stream>>>(kbuf, vbuf, kmax, kinv, ctx);
    k_w2    <<<B_ * H_,           256, 0, stream>>>(wout, ctx, W2);
    k_out   <<<dim3(L_ / LT, B_), 256, 0, stream>>>(W2, qbuf, bout, out, parts);
    k_gnstat<<<B_,                256, 0, stream>>>(parts, gns);
    k_gnorm <<<(int)(PLANE / 4 / 256), 256, 0, stream>>>(out, gns, gnw, gnb);
}